// GIBGAT_4071628996669
// MI455X (gfx1250) — compile-verified
//
#include <hip/hip_runtime.h>
#include <math.h>

// ---------------------------------------------------------------------------
// Two-layer GAT (PyG-style, aggr='add') with reparam head, MI455X / gfx1250.
// Only the mean path is live (ixz / softplus / KL are discarded by the
// reference), so we compute:
//   h1raw = x @ W1            (fp32 WMMA 16x16x4, LDS-staged B panel via
//                              global_load_async_to_lds_b128)
//   alpha via per-node att dots + edge softmax (atomic max / atomic add)
//   out1  = scatter_add(alpha * h1raw[src]) + bias1 ; h1 = elu(mean half)
//   h2raw = h1 @ W2           (same WMMA kernel, NT=2)
//   ... same attention, heads=1 ...      -> out = mean half + bias2
// ---------------------------------------------------------------------------

typedef __attribute__((ext_vector_type(2))) float v2f;
typedef __attribute__((ext_vector_type(8))) float v8f;

#define NEG_SLOPE 0.2f

// ---- gfx1250 async copy: global -> LDS, 16B per lane, ASYNCcnt-tracked ----
__device__ __forceinline__ void async_g2lds_b128(unsigned lds_off,
                                                 const void* gaddr)
{
    unsigned long long ga = (unsigned long long)(uintptr_t)gaddr;
    asm volatile("global_load_async_to_lds_b128 %0, %1, off"
                 :: "v"(lds_off), "v"(ga) : "memory");
}
__device__ __forceinline__ void wait_asynccnt0()
{
    asm volatile("s_wait_asynccnt 0" ::: "memory");
}

// ---------------------------------------------------------------------------
// fp32 WMMA GEMM: C[M,N] = A[M,K] * B[K,N], all row-major. K == 256 fixed.
// Block = 256 threads = 8 waves. Each block computes a 128 x (16*NT) tile:
//   - the K x (16*NT) B panel is staged in LDS once per block via
//     GLOBAL_LOAD_ASYNC_TO_LDS_B128 (no VGPR round-trip),
//   - each wave register-blocks NT 16x16 accumulators over one 16-row band,
//     so one b64 A load feeds NT V_WMMA_F32_16X16X4_F32 ops.
// Requires N % (16*NT) == 0 (512 % 64, 32 % 32: both hold); M is guarded.
//
// WMMA operand layouts (ISA 7.12.2):
//   A 16x4 : lanes 0-15 hold (M=lane, K=0/1 in v0/v1); lanes 16-31 hold
//            (M=lane-16, K=2/3)  -> one contiguous b64 load per lane.
//   B 4x16 : v0/v1 hold rows K+koff / K+koff+1, N striped across 16 lanes.
//   C/D    : VGPR r -> (M=r, N=lane) lanes 0-15, (M=r+8, N=lane-16) else.
// ---------------------------------------------------------------------------
template <int NT>
__global__ void __launch_bounds__(256)
wmma_gemm_f32_lds(const float* __restrict__ A, const float* __restrict__ B,
                  float* __restrict__ C, int M, int N)
{
    constexpr int K     = 256;
    constexpr int NCOLS = 16 * NT;
    __shared__ float Bs[K * NCOLS];          // NT=4: 64KB, NT=2: 32KB (of 320KB/WGP)

    const int nBlocks = N / NCOLS;
    const int mBase = (blockIdx.x / nBlocks) * 128;
    const int n0    = (blockIdx.x % nBlocks) * NCOLS;
    const int tid   = threadIdx.x;

    // ---- stage B panel: K x NCOLS floats, 16B chunks, async to LDS ----
    const unsigned bsBase = (unsigned)(uintptr_t)&Bs[0];   // LDS aperture: low 32b
    constexpr int CHUNKS = (K * NCOLS) / 4;                // 16B chunks
    for (int i = tid; i < CHUNKS; i += 256) {
        const int el = i * 4;
        const int kr = el / NCOLS;
        const int nc = el % NCOLS;
        async_g2lds_b128(bsBase + (unsigned)(i * 16),
                         B + (size_t)kr * N + n0 + nc);
    }
    wait_asynccnt0();
    __syncthreads();

    // ---- per-wave 16 x NCOLS strip ----
    const int waveId = tid >> 5;
    const int lane   = tid & 31;
    const int half   = lane >> 4;
    const int l16    = lane & 15;

    int arow = mBase + waveId * 16 + l16;
    if (arow >= M) arow = M - 1;                 // clamp; stores are guarded
    const float* aptr = A + (size_t)arow * K + half * 2;   // 8B aligned

    v8f acc[NT];
#pragma unroll
    for (int t = 0; t < NT; ++t)
        acc[t] = (v8f){0.f, 0.f, 0.f, 0.f, 0.f, 0.f, 0.f, 0.f};

    for (int k = 0; k < K; k += 4) {
        const v2f a = *(const v2f*)(aptr + k);   // b64: K=k+{0,1} / k+{2,3}
        const int kb = (k + half * 2) * NCOLS;
#pragma unroll
        for (int t = 0; t < NT; ++t) {
            v2f b;
            b.x = Bs[kb + t * 16 + l16];         // row k+koff
            b.y = Bs[kb + NCOLS + t * 16 + l16]; // row k+koff+1
            acc[t] = __builtin_amdgcn_wmma_f32_16x16x4_f32(
                /*neg_a=*/false, a, /*neg_b=*/false, b,
                /*c_mod=*/(short)0, acc[t], /*reuse_a=*/false, /*reuse_b=*/false);
        }
    }

#pragma unroll
    for (int r = 0; r < 8; ++r) {
        const int row = mBase + waveId * 16 + r + half * 8;
        if (row < M) {
#pragma unroll
            for (int t = 0; t < NT; ++t)
                C[(size_t)row * N + n0 + t * 16 + l16] = acc[t][r];
        }
    }
}

// ---------------------------------------------------------------------------
// Zero a float buffer (accumulators).
// ---------------------------------------------------------------------------
__global__ void zero_f32(float* __restrict__ p, long long n)
{
    long long i = blockIdx.x * (long long)blockDim.x + threadIdx.x;
    if (i < n) p[i] = 0.f;
}

// ---------------------------------------------------------------------------
// Per-(node, head) attention dot products; also initializes lmax/denom.
//   ai[n,h] = sum_c h[n,h,c] * att_i[h,c]   (same for aj / att_j)
// ---------------------------------------------------------------------------
__global__ void node_att_dots(const float* __restrict__ h,
                              const float* __restrict__ att_i,
                              const float* __restrict__ att_j,
                              float* __restrict__ ai, float* __restrict__ aj,
                              float* __restrict__ lmax, float* __restrict__ denom,
                              int N, int H, int C2)
{
    int idx = blockIdx.x * blockDim.x + threadIdx.x;
    if (idx >= N * H) return;
    const int n  = idx / H;
    const int hh = idx % H;
    const float* row = h + ((size_t)n * H + hh) * C2;
    const float* wi  = att_i + hh * C2;
    const float* wj  = att_j + hh * C2;
    float si = 0.f, sj = 0.f;
    for (int c = 0; c < C2; ++c) {
        const float v = row[c];
        si = fmaf(v, wi[c], si);
        sj = fmaf(v, wj[c], sj);
    }
    ai[idx] = si;
    aj[idx] = sj;
    lmax[idx]  = -3.402823466e38f;   // segment-max identity
    denom[idx] = 0.f;
}

// ---------------------------------------------------------------------------
// Pass 1 over edges: logits -> leaky_relu -> atomic segment max over dst.
// ---------------------------------------------------------------------------
__global__ void edge_logit_max(const int* __restrict__ src, const int* __restrict__ dst,
                               const float* __restrict__ ai, const float* __restrict__ aj,
                               float* __restrict__ lmax, long long EH, int H)
{
    long long idx = blockIdx.x * (long long)blockDim.x + threadIdx.x;
    if (idx >= EH) return;
    const int e  = (int)(idx / H);
    const int hh = (int)(idx % H);
    const int s = src[e], d = dst[e];
    float lg = ai[(size_t)d * H + hh] + aj[(size_t)s * H + hh];
    lg = lg > 0.f ? lg : NEG_SLOPE * lg;
    atomicMax(&lmax[(size_t)d * H + hh], lg);   // global_atomic_max_num_f32
}

// ---------------------------------------------------------------------------
// Pass 2 over edges: ex = exp(logit - lmax[dst]); denom[dst] += ex.
// ---------------------------------------------------------------------------
__global__ void edge_exp_sum(const int* __restrict__ src, const int* __restrict__ dst,
                             const float* __restrict__ ai, const float* __restrict__ aj,
                             const float* __restrict__ lmax,
                             float* __restrict__ ex, float* __restrict__ denom,
                             long long EH, int H)
{
    long long idx = blockIdx.x * (long long)blockDim.x + threadIdx.x;
    if (idx >= EH) return;
    const int e  = (int)(idx / H);
    const int hh = (int)(idx % H);
    const int s = src[e], d = dst[e];
    float lg = ai[(size_t)d * H + hh] + aj[(size_t)s * H + hh];
    lg = lg > 0.f ? lg : NEG_SLOPE * lg;
    const float v = __expf(lg - lmax[(size_t)d * H + hh]);
    ex[idx] = v;
    atomicAdd(&denom[(size_t)d * H + hh], v);
}

// ---------------------------------------------------------------------------
// Pass 3: weighted scatter.  One thread per (edge, channel); consecutive
// threads hit consecutive channels -> coalesced loads of h[src,*] and
// contiguous atomics into out[dst,*] (all L2-resident on MI455X's 192MB L2).
// ---------------------------------------------------------------------------
__global__ void scatter_weighted(const int* __restrict__ src, const int* __restrict__ dst,
                                 const float* __restrict__ ex, const float* __restrict__ denom,
                                 const float* __restrict__ h, float* __restrict__ out,
                                 long long total, int E, int H, int C2)
{
    const int CH = H * C2;
    long long gid = blockIdx.x * (long long)blockDim.x + threadIdx.x;
    if (gid >= total) return;
    const int e   = (int)(gid / CH);
    const int idx = (int)(gid % CH);
    const int hh  = idx / C2;
    const int s = src[e], d = dst[e];
    if (e + 4 < E) {
        // warm L2 for an upcoming edge's source row (global_prefetch_b8)
        __builtin_prefetch(&h[(size_t)src[e + 4] * CH + idx], 0, 1);
    }
    const float alpha = ex[(size_t)e * H + hh] / denom[(size_t)d * H + hh];
    atomicAdd(&out[(size_t)d * CH + idx], alpha * h[(size_t)s * CH + idx]);
}

// ---------------------------------------------------------------------------
// Layer-1 finalize: mean half + bias, then ELU (std/softplus/KL are dead).
//   h1[n, h*C + c] = elu(acc[n, h*C2 + c] + bias[h*C2 + c]),  C = C2/2
// ---------------------------------------------------------------------------
__global__ void finalize_elu_mean(const float* __restrict__ acc,
                                  const float* __restrict__ bias,
                                  float* __restrict__ h1, int N, int H, int C2)
{
    const int C = C2 >> 1;
    int idx = blockIdx.x * blockDim.x + threadIdx.x;
    if (idx >= N * H * C) return;
    const int n  = idx / (H * C);
    const int r  = idx % (H * C);
    const int hh = r / C;
    const int c  = r % C;
    const float m = acc[((size_t)n * H + hh) * C2 + c] + bias[hh * C2 + c];
    h1[((size_t)n * H + hh) * C + c] = m > 0.f ? m : (__expf(m) - 1.f);
}

// ---------------------------------------------------------------------------
// Layer-2 finalize: output = mean half + bias (no activation).
// ---------------------------------------------------------------------------
__global__ void finalize_mean(const float* __restrict__ acc,
                              const float* __restrict__ bias,
                              float* __restrict__ out, int N, int C2)
{
    const int C = C2 >> 1;
    int idx = blockIdx.x * blockDim.x + threadIdx.x;
    if (idx >= N * C) return;
    const int n = idx / C;
    const int c = idx % C;
    out[(size_t)n * C + c] = acc[(size_t)n * C2 + c] + bias[c];
}

// ---------------------------------------------------------------------------
// Host-side orchestration.
// ---------------------------------------------------------------------------
static inline unsigned grid1(long long total, int block)
{
    return (unsigned)((total + block - 1) / block);
}

extern "C" void kernel_launch(void* const* d_in, const int* in_sizes, int n_in,
                              void* d_out, int out_size, void* d_ws, size_t ws_size,
                              hipStream_t stream)
{
    const float* x      = (const float*)d_in[0];
    const int*   eidx   = (const int*)  d_in[1];
    const float* W1     = (const float*)d_in[2];
    const float* att_i1 = (const float*)d_in[3];
    const float* att_j1 = (const float*)d_in[4];
    const float* bias1  = (const float*)d_in[5];
    const float* W2     = (const float*)d_in[6];
    const float* att_i2 = (const float*)d_in[7];
    const float* att_j2 = (const float*)d_in[8];
    const float* bias2  = (const float*)d_in[9];
    float* out = (float*)d_out;

    const int F1 = 256;
    const int N  = in_sizes[0] / F1;       // 20000
    const int E  = in_sizes[1] / 2;        // 320000
    const int H1 = 8,  C2a = 64;           // layer 1: heads=8, c2=64
    const int H2 = 1,  C2b = 32;           // layer 2: heads=1, c2=32
    const int CH1 = H1 * C2a;              // 512
    const int CH2 = H2 * C2b;              // 32

    const int* src = eidx;                 // edge_index[0]
    const int* dst = eidx + E;             // edge_index[1]

    // ---- workspace carve-up (all fp32) ----
    char* ws = (char*)d_ws;
    size_t off = 0;
    auto carve = [&](size_t nElem) { float* p = (float*)(ws + off); off += nElem * sizeof(float); return p; };
    float* h1raw   = carve((size_t)N * CH1);   // 40.96 MB
    float* out1acc = carve((size_t)N * CH1);   // 40.96 MB
    float* h1      = carve((size_t)N * H1 * (C2a / 2)); // 20.48 MB
    float* ai1     = carve((size_t)N * H1);
    float* aj1     = carve((size_t)N * H1);
    float* lmax1   = carve((size_t)N * H1);
    float* den1    = carve((size_t)N * H1);
    float* ex1     = carve((size_t)E * H1);    // 10.24 MB
    float* h2raw   = carve((size_t)N * CH2);
    float* out2acc = carve((size_t)N * CH2);
    float* ai2     = carve((size_t)N * H2);
    float* aj2     = carve((size_t)N * H2);
    float* lmax2   = carve((size_t)N * H2);
    float* den2    = carve((size_t)N * H2);
    float* ex2     = carve((size_t)E * H2);
    (void)ws_size; (void)n_in; (void)out_size;

    const int B = 256;
    const int mBlocks = (N + 127) / 128;   // 157 (last block partially guarded)

    // ================= Layer 1 =================
    zero_f32<<<grid1((long long)N * CH1, B), B, 0, stream>>>(out1acc, (long long)N * CH1);

    // h1raw = x @ W1   (20000x256 @ 256x512), fp32 WMMA, NT=4 (64-wide panel)
    wmma_gemm_f32_lds<4><<<mBlocks * (CH1 / 64), 256, 0, stream>>>(x, W1, h1raw, N, CH1);

    node_att_dots<<<grid1((long long)N * H1, B), B, 0, stream>>>(
        h1raw, att_i1, att_j1, ai1, aj1, lmax1, den1, N, H1, C2a);

    edge_logit_max<<<grid1((long long)E * H1, B), B, 0, stream>>>(
        src, dst, ai1, aj1, lmax1, (long long)E * H1, H1);

    edge_exp_sum<<<grid1((long long)E * H1, B), B, 0, stream>>>(
        src, dst, ai1, aj1, lmax1, ex1, den1, (long long)E * H1, H1);

    scatter_weighted<<<grid1((long long)E * CH1, B), B, 0, stream>>>(
        src, dst, ex1, den1, h1raw, out1acc, (long long)E * CH1, E, H1, C2a);

    finalize_elu_mean<<<grid1((long long)N * H1 * (C2a / 2), B), B, 0, stream>>>(
        out1acc, bias1, h1, N, H1, C2a);

    // ================= Layer 2 =================
    zero_f32<<<grid1((long long)N * CH2, B), B, 0, stream>>>(out2acc, (long long)N * CH2);

    // h2raw = h1 @ W2  (20000x256 @ 256x32), fp32 WMMA, NT=2 (32-wide panel)
    wmma_gemm_f32_lds<2><<<mBlocks * (CH2 / 32), 256, 0, stream>>>(h1, W2, h2raw, N, CH2);

    node_att_dots<<<grid1((long long)N * H2, B), B, 0, stream>>>(
        h2raw, att_i2, att_j2, ai2, aj2, lmax2, den2, N, H2, C2b);

    edge_logit_max<<<grid1((long long)E * H2, B), B, 0, stream>>>(
        src, dst, ai2, aj2, lmax2, (long long)E * H2, H2);

    edge_exp_sum<<<grid1((long long)E * H2, B), B, 0, stream>>>(
        src, dst, ai2, aj2, lmax2, ex2, den2, (long long)E * H2, H2);

    scatter_weighted<<<grid1((long long)E * CH2, B), B, 0, stream>>>(
        src, dst, ex2, den2, h2raw, out2acc, (long long)E * CH2, E, H2, C2b);

    finalize_mean<<<grid1((long long)N * (C2b / 2), B), B, 0, stream>>>(
        out2acc, bias2, out, N, C2b);
}